// MischiefGNN_10771777978427
// MI455X (gfx1250) — compile-verified
//
#include <hip/hip_runtime.h>
#include <hip/hip_bf16.h>

// ---------------------------------------------------------------------------
// MischiefGNN for MI455X (gfx1250, wave32)
//
//  * Graphs processed sequentially: ~6.5 MB per-graph scratch stays resident
//    in the 192 MB L2, so edge scatters run at L2 atomic throughput.
//  * SAGEConv2 + mean-pool is linear => folded into two 64x64 matvecs on the
//    pooled vectors (mean(agg2/deg), mean(h)).
//  * SAGEConv1 runs on V_WMMA_F32_16X16X4_F32. Inputs pre-padded K:15->16 in
//    workspace so the WMMA inner loop is branch-free (no exec-mask churn).
// ---------------------------------------------------------------------------

#define TT 32
#define NN 10000
#define EE 160000
#define IND 15
#define HH 64

typedef __attribute__((ext_vector_type(2))) float v2f;
typedef __attribute__((ext_vector_type(8))) float v8f;

// ---------------- zero scratch (deg | agg1 | agg2 | sums, contiguous) -------
__global__ void zero_kernel(float* __restrict__ p, int n) {
    int i = blockIdx.x * blockDim.x + threadIdx.x;
    int stride = gridDim.x * blockDim.x;
    for (; i < n; i += stride) p[i] = 0.0f;
}

// ---------------- pad weights [15,64] -> [16,64] (once per call) -----------
__global__ void pad_w_kernel(const float* __restrict__ w1l,
                             const float* __restrict__ w1r,
                             float* __restrict__ w1lp,   // [16,64]
                             float* __restrict__ w1rp) { // [16,64]
    int i = blockIdx.x * blockDim.x + threadIdx.x;   // 0..2047
    if (i >= 2 * 16 * HH) return;
    int m = i >> 10;          // 0 = lin_l, 1 = lin_r
    int j = i & 1023;
    int k = j >> 6;
    int c = j & 63;
    float v = (k < IND) ? (m ? w1r[k * HH + c] : w1l[k * HH + c]) : 0.0f;
    if (m) w1rp[j] = v; else w1lp[j] = v;
}

// ---------------- pad node features [N,15] -> [N,16] (per graph) -----------
__global__ void pad_x_kernel(const float* __restrict__ xg,  // [N,15]
                             float* __restrict__ x16) {     // [N,16]
    int i = blockIdx.x * blockDim.x + threadIdx.x;
    if (i >= NN * 16) return;
    int n = i >> 4;
    int k = i & 15;
    x16[i] = (k < IND) ? xg[n * IND + k] : 0.0f;
}

// ---------------- edge scatter for layer 1 + degree count ------------------
// thread i -> edge e = i/16, channel ch = i%16 ; ch<15 scatters x, ch==15 deg
__global__ void scatter1_kernel(const int* __restrict__ src,
                                const int* __restrict__ dst,
                                const float* __restrict__ xg,   // [N,15]
                                float* __restrict__ agg1,       // [N,16] padded
                                float* __restrict__ deg) {      // [N]
    int i = blockIdx.x * blockDim.x + threadIdx.x;
    if (i >= EE * 16) return;
    int e  = i >> 4;
    int ch = i & 15;
    int d  = dst[e];
    if (ch < IND) {
        atomicAdd(&agg1[d * 16 + ch], xg[src[e] * IND + ch]);
    } else {
        atomicAdd(&deg[d], 1.0f);
    }
}

// ---------------- layer-1 GEMM on WMMA fp32 --------------------------------
// h[N,64] = relu( (agg1/deg) @ w1_l + x @ w1_r + b1 )
// One wave per 16x16 tile; all operands pre-padded to K=16 => branch-free.
__global__ void gemm1_wmma_kernel(const float* __restrict__ x16,   // [N,16]
                                  const float* __restrict__ agg1,  // [N,16]
                                  const float* __restrict__ deg,   // [N]
                                  const float* __restrict__ w1lp,  // [16,64]
                                  const float* __restrict__ w1rp,  // [16,64]
                                  const float* __restrict__ b1,    // [64]
                                  float* __restrict__ h) {         // [N,64]
    const int wave = threadIdx.x >> 5;
    const int lane = threadIdx.x & 31;
    const int tile = blockIdx.x * (blockDim.x >> 5) + wave;  // 0..2499
    const int tr   = tile >> 2;   // 0..624 row tile
    const int tc   = tile & 3;    // 0..3   col tile
    const int half = lane >> 4;   // selects K pair within chunk (ISA A layout)
    const int l16  = lane & 15;
    const int row  = tr * 16 + l16;   // A-matrix row for this lane
    const int col  = tc * 16 + l16;   // B/C column for this lane

    const float dinv = 1.0f / fmaxf(deg[row], 1.0f);

    v8f acc = {};
#pragma unroll
    for (int kc = 0; kc < 4; ++kc) {
        const int k0 = kc * 4 + half * 2;   // 0,2,...,14 per half-wave
        // neighbor aggregate term: A = (agg1/deg), aligned float2 load
        v2f a1 = *(const v2f*)(agg1 + row * 16 + k0);
        a1 *= dinv;
        v2f b1v; b1v.x = w1lp[k0 * HH + col];
                 b1v.y = w1lp[(k0 + 1) * HH + col];
        acc = __builtin_amdgcn_wmma_f32_16x16x4_f32(false, a1, false, b1v,
                                                    (short)0, acc, false, false);
        // root term: A = x (padded), aligned float2 load
        v2f a2 = *(const v2f*)(x16 + row * 16 + k0);
        v2f b2v; b2v.x = w1rp[k0 * HH + col];
                 b2v.y = w1rp[(k0 + 1) * HH + col];
        acc = __builtin_amdgcn_wmma_f32_16x16x4_f32(false, a2, false, b2v,
                                                    (short)0, acc, false, false);
    }

    const float bias = b1[col];
#pragma unroll
    for (int v = 0; v < 8; ++v) {
        int r = tr * 16 + v + half * 8;     // C/D layout: VGPR v, half selects M+8
        h[r * HH + col] = fmaxf(acc[v] + bias, 0.0f);
    }
}

// ---------------- edge scatter for layer 2 ---------------------------------
// thread i -> edge e = i/32, channel pair ch = 2*(i%32); float2 gather + 2 atomics
__global__ void scatter2_kernel(const int* __restrict__ src,
                                const int* __restrict__ dst,
                                const float* __restrict__ h,     // [N,64]
                                float* __restrict__ agg2) {      // [N,64]
    int i = blockIdx.x * blockDim.x + threadIdx.x;
    if (i >= EE * 32) return;
    int e  = i >> 5;
    int ch = (i & 31) << 1;
    v2f v = *(const v2f*)(h + src[e] * HH + ch);
    float* p = agg2 + dst[e] * HH + ch;
    atomicAdd(p,     v.x);
    atomicAdd(p + 1, v.y);
}

// ---------------- pooled reduction -----------------------------------------
// sums[c]     = sum_n agg2[n][c] / max(deg[n],1)
// sums[64+c]  = sum_n h[n][c]
__global__ void reduce_pool_kernel(const float* __restrict__ h,
                                   const float* __restrict__ agg2,
                                   const float* __restrict__ deg,
                                   float* __restrict__ sums) {
    int c  = threadIdx.x & 63;
    int r0 = blockIdx.x * (blockDim.x >> 6) + (threadIdx.x >> 6);
    int rs = gridDim.x * (blockDim.x >> 6);
    float sa = 0.0f, sh = 0.0f;
    for (int n = r0; n < NN; n += rs) {
        float dinv = 1.0f / fmaxf(deg[n], 1.0f);
        sa += agg2[n * HH + c] * dinv;
        sh += h[n * HH + c];
    }
    atomicAdd(&sums[c], sa);
    atomicAdd(&sums[HH + c], sh);
}

// ---------------- finish layer 2 via linearity through the mean ------------
// seq_t[c] = b2[c] + (sumsA/N) @ w2_l[:,c] + (sumsH/N) @ w2_r[:,c]
__global__ void finish_seq_kernel(const float* __restrict__ sums,
                                  const float* __restrict__ w2l,
                                  const float* __restrict__ w2r,
                                  const float* __restrict__ b2,
                                  float* __restrict__ seq_t) {
    int c = threadIdx.x;   // 64 threads
    const float invN = 1.0f / (float)NN;
    float acc = b2[c];
    for (int k = 0; k < HH; ++k) {
        acc += sums[k]      * invN * w2l[k * HH + c];
        acc += sums[HH + k] * invN * w2r[k * HH + c];
    }
    seq_t[c] = acc;
}

// ---------------- GRU over T steps + classifier head -----------------------
__global__ void gru_head_kernel(const float* __restrict__ seq,   // [T,64]
                                const float* __restrict__ w_ih,  // [192,64]
                                const float* __restrict__ w_hh,  // [192,64]
                                const float* __restrict__ b_ih,  // [192]
                                const float* __restrict__ b_hh,  // [192]
                                const float* __restrict__ wc1,   // [64,32]
                                const float* __restrict__ bc1,   // [32]
                                const float* __restrict__ wc2,   // [32,3]
                                const float* __restrict__ bc2,   // [3]
                                float* __restrict__ out) {       // [3]
    __shared__ float hs[HH];
    __shared__ float hid[32];
    int c = threadIdx.x;       // blockDim == 64
    hs[c] = 0.0f;
    __syncthreads();
    for (int t = 0; t < TT; ++t) {
        const float* xt = seq + t * HH;
        float ir = b_ih[c], iz = b_ih[HH + c], in_ = b_ih[2 * HH + c];
        float hr = b_hh[c], hz = b_hh[HH + c], hn  = b_hh[2 * HH + c];
        for (int k = 0; k < HH; ++k) {
            float xv = xt[k], hv = hs[k];
            ir  += w_ih[c * HH + k] * xv;
            iz  += w_ih[(HH + c) * HH + k] * xv;
            in_ += w_ih[(2 * HH + c) * HH + k] * xv;
            hr  += w_hh[c * HH + k] * hv;
            hz  += w_hh[(HH + c) * HH + k] * hv;
            hn  += w_hh[(2 * HH + c) * HH + k] * hv;
        }
        float r = 1.0f / (1.0f + __expf(-(ir + hr)));
        float z = 1.0f / (1.0f + __expf(-(iz + hz)));
        float n = tanhf(in_ + r * hn);
        float hnew = (1.0f - z) * n + z * hs[c];
        __syncthreads();
        hs[c] = hnew;
        __syncthreads();
    }
    if (c < 32) {
        float a = bc1[c];
        for (int k = 0; k < HH; ++k) a += hs[k] * wc1[k * 32 + c];
        hid[c] = fmaxf(a, 0.0f);
    }
    __syncthreads();
    if (c < 3) {
        float a = bc2[c];
        for (int j = 0; j < 32; ++j) a += hid[j] * wc2[j * 3 + c];
        out[c] = a;
    }
}

// ---------------------------------------------------------------------------
extern "C" void kernel_launch(void* const* d_in, const int* in_sizes, int n_in,
                              void* d_out, int out_size, void* d_ws, size_t ws_size,
                              hipStream_t stream) {
    const float* x    = (const float*)d_in[0];   // [T,N,15]
    const int*   ei   = (const int*)  d_in[1];   // [T,2,E]
    const float* w1l  = (const float*)d_in[2];   // [15,64]
    const float* b1   = (const float*)d_in[3];   // [64]
    const float* w1r  = (const float*)d_in[4];   // [15,64]
    const float* w2l  = (const float*)d_in[5];   // [64,64]
    const float* b2   = (const float*)d_in[6];   // [64]
    const float* w2r  = (const float*)d_in[7];   // [64,64]
    const float* w_ih = (const float*)d_in[8];   // [192,64]
    const float* w_hh = (const float*)d_in[9];   // [192,64]
    const float* b_ih = (const float*)d_in[10];  // [192]
    const float* b_hh = (const float*)d_in[11];  // [192]
    const float* wc1  = (const float*)d_in[12];  // [64,32]
    const float* bc1  = (const float*)d_in[13];  // [32]
    const float* wc2  = (const float*)d_in[14];  // [32,3]
    const float* bc2  = (const float*)d_in[15];  // [3]
    float* out = (float*)d_out;

    // Scratch layout (floats). Zeroed region [deg..sums] is contiguous.
    float* ws    = (float*)d_ws;
    float* deg   = ws;                          // N
    float* agg1  = deg  + NN;                   // N*16 (K padded, col 15 stays 0)
    float* agg2  = agg1 + NN * 16;              // N*64
    float* sums  = agg2 + NN * HH;              // 128
    float* hbuf  = sums + 128;                  // N*64
    float* x16   = hbuf + NN * HH;              // N*16 padded features
    float* w1lp  = x16  + NN * 16;              // 16*64
    float* w1rp  = w1lp + 16 * HH;              // 16*64
    float* seq   = w1rp + 16 * HH;              // T*64
    const size_t needed = (size_t)(NN + NN * 16 + NN * HH + 128 + NN * HH +
                                   NN * 16 + 2 * 16 * HH + TT * HH) * sizeof(float);
    if (ws_size < needed) return;               // deterministic no-op guard

    const int zeroN = NN + NN * 16 + NN * HH + 128;   // deg+agg1+agg2+sums
    const int s1thr = EE * 16;
    const int s2thr = EE * 32;                        // float2 per thread
    const int pxthr = NN * 16;

    pad_w_kernel<<<8, 256, 0, stream>>>(w1l, w1r, w1lp, w1rp);

    for (int t = 0; t < TT; ++t) {
        const float* xg  = x  + (size_t)t * NN * IND;
        const int*   src = ei + (size_t)t * 2 * EE;
        const int*   dst = src + EE;

        zero_kernel<<<2048, 256, 0, stream>>>(deg, zeroN);
        pad_x_kernel<<<(pxthr + 255) / 256, 256, 0, stream>>>(xg, x16);
        scatter1_kernel<<<(s1thr + 255) / 256, 256, 0, stream>>>(src, dst, xg, agg1, deg);
        // 2500 tiles, 4 waves (128 threads) per block -> 625 blocks, no idle waves
        gemm1_wmma_kernel<<<625, 128, 0, stream>>>(x16, agg1, deg, w1lp, w1rp, b1, hbuf);
        scatter2_kernel<<<(s2thr + 255) / 256, 256, 0, stream>>>(src, dst, hbuf, agg2);
        reduce_pool_kernel<<<64, 256, 0, stream>>>(hbuf, agg2, deg, sums);
        finish_seq_kernel<<<1, 64, 0, stream>>>(sums, w2l, w2r, b2, seq + t * HH);
    }
    gru_head_kernel<<<1, 64, 0, stream>>>(seq, w_ih, w_hh, b_ih, b_hh,
                                          wc1, bc1, wc2, bc2, out);
}